// PatchletsExtractorStrided_83743272337984
// MI455X (gfx1250) — compile-verified
//
#include <hip/hip_runtime.h>

// ---------------------------------------------------------------------------
// KNN point-tracking (PatchletsExtractorStrided) for gfx1250 / MI455X.
//
// Distance tiles via V_WMMA_F32_16X16X4_F32:
//   d2[m][n] = |q_m|^2 + |p_n|^2 - 2 q_m . p_n
//   A row m  = (-2qx, -2qy, -2qz, 1)        (16x4, K padded to 4)
//   B col n  = ( px ,  py ,  pz , |p|^2)    (4x16)
//   C[m][n]  = |q_m|^2
//
// One wave owns 32 queries of one (b, seg, dir) chain and runs all 8
// sequential tracking steps. All 8 waves of a block share the same chain,
// so the 24 KB db frame is staged once per step with the Tensor Data Mover
// (tensor_load_to_lds, TENSORcnt) into a double-buffered LDS region.
// ---------------------------------------------------------------------------

#define USE_TDM  1

#define KNN_K    16
#define N_PTS    2048
#define T_FRAMES 32
#define BATCH    8
#define STRIDE_T 8
#define SEL      (N_PTS / 2)
#define N_TILES  (N_PTS / 16)
#define FRAME_F  (N_PTS * 3)          // floats per frame (6144)

typedef float v2f __attribute__((ext_vector_type(2)));
typedef float v8f __attribute__((ext_vector_type(8)));
typedef unsigned int v4u __attribute__((ext_vector_type(4)));
typedef unsigned int v8u __attribute__((ext_vector_type(8)));

#if defined(__has_builtin)
#if __has_builtin(__builtin_amdgcn_wmma_f32_16x16x4_f32)
#define HAVE_WMMA_F32X4 1
#endif
#if __has_builtin(__builtin_amdgcn_s_wait_tensorcnt)
#define HAVE_WAIT_TENSORCNT 1
#endif
#endif

// Sorted-ascending top-K insertion network (registers only, fully unrolled).
#define KNN_INSERT(dvx, dix)                                        \
  if ((dvx) < best_d[KNN_K - 1]) {                                  \
    float v_ = (dvx); int vi_ = (dix);                              \
    _Pragma("unroll")                                               \
    for (int s_ = 0; s_ < KNN_K; ++s_) {                            \
      bool  lt_ = v_ < best_d[s_];                                  \
      float nd_ = lt_ ? v_        : best_d[s_];                     \
      int   ni_ = lt_ ? vi_       : best_i[s_];                     \
      float od_ = lt_ ? best_d[s_] : v_;                            \
      int   oi_ = lt_ ? best_i[s_] : vi_;                           \
      best_d[s_] = nd_; best_i[s_] = ni_;                           \
      v_ = od_; vi_ = oi_;                                          \
    }                                                               \
  }

#if USE_TDM
// Issue a TDM DMA of one full frame (6144 dwords) into LDS at lds_bytes_off.
// D# group0/group1 built per CDNA5 ISA (2D: VADDR2/3 NULL).
__device__ __forceinline__ void tdm_load_frame(const float* gsrc,
                                               unsigned lds_bytes_off) {
  const unsigned long long ga = (unsigned long long)(size_t)gsrc;
  v4u g0;
  g0[0] = 1u;                                            // count=1, user mode
  g0[1] = lds_bytes_off;                                 // lds_addr
  g0[2] = (unsigned)(ga & 0xFFFFFFFFu);                  // global_addr[31:0]
  g0[3] = (unsigned)((ga >> 32) & 0x1FFFFFFu)            // global_addr[56:32]
        | (2u << 30);                                    // type=2 (image)
  v8u g1;
  g1[0] = 2u << 16;                                      // data_size=4B, no mcast
  g1[1] = (FRAME_F & 0xFFFFu) << 16;                     // tensor_dim0 lo16
  g1[2] = ((unsigned)FRAME_F >> 16) | (1u << 16);        // dim0 hi16 | dim1=1 lo16
  g1[3] = ((unsigned)FRAME_F) << 16;                     // dim1 hi16=0 | tile_dim0
  g1[4] = 1u;                                            // tile_dim1=1, tile_dim2=0
  g1[5] = (unsigned)FRAME_F;                             // dim0_stride lo32
  g1[6] = 0u;                                            // dim0_stride hi | dim1_stride lo
  g1[7] = 0u;                                            // dim1_stride hi
  asm volatile("tensor_load_to_lds %0, %1" :: "s"(g0), "s"(g1) : "memory");
}
#endif

__global__ __launch_bounds__(256)
void knn_track_kernel(const float* __restrict__ pts_in,
                      float* __restrict__ out_idx0,
                      float* __restrict__ out_dist,
                      float* __restrict__ out_idx2,
                      float* __restrict__ out_pts,
                      float* __restrict__ out_feats,
                      float* __restrict__ out_outx)
{
#if USE_TDM
  __shared__ float lds_frame[2][FRAME_F];   // 2 x 24 KB double-buffered frame
#endif
  __shared__ float lds_tiles[8][512];       // 2 x 16x16 f32 tiles per wave

  const int lane  = threadIdx.x & 31;
  const int wib   = threadIdx.x >> 5;                 // wave-in-block (0..7)
  const int wid   = blockIdx.x * 8 + wib;             // 0..4095
  const int qtile = wid & 63;                         // 64 query tiles of 32
  const int dir   = (wid >> 6) & 1;                   // 0=fwd, 1=flipped
  const int seg   = (wid >> 7) & 3;                   // 4 segments
  const int b     = wid >> 9;                         // batch
  const int qg    = qtile * 32 + lane;                // global query index
  const int t0    = seg * STRIDE_T;
  const bool hi   = lane >= 16;

  const float* batch_base = pts_in + (size_t)b * T_FRAMES * N_PTS * 3;

  // Initial carry = first frame of (possibly flipped) segment.
  float qx, qy, qz, qq;
  {
    const int f0 = dir ? (t0 + 7) : t0;
    const float* p0 = batch_base + ((size_t)f0 * N_PTS + qg) * 3;
    qx = p0[0]; qy = p0[1]; qz = p0[2];
    qq = qx * qx + qy * qy + qz * qz;
  }

  for (int j = 0; j < STRIDE_T; ++j) {
    const int f_db  = dir ? (t0 + 7 - j) : (t0 + j);
    const int f_ft  = dir ? ((j == 0) ? (t0 + 7) : (t0 + 8 - j))
                          : ((j == 0) ? t0       : (t0 + j - 1));
    const int t_out = dir ? (t0 + 7 - j) : (t0 + j);

    const float* fbg = batch_base + (size_t)f_db * N_PTS * 3;  // db frame (global)
    const float* ff  = batch_base + (size_t)f_ft * N_PTS * 3;  // feature frame

#if USE_TDM
    // Stage the shared db frame into LDS via the Tensor Data Mover.
    // Double buffer + one barrier per step: buffer j&1 was last read in step
    // j-2, and every wave passed step j-1's barrier after that.
    const int buf = j & 1;
    if (wib == 0) {
      tdm_load_frame(fbg, (unsigned)(size_t)(&lds_frame[buf][0]));
#ifdef HAVE_WAIT_TENSORCNT
      __builtin_amdgcn_s_wait_tensorcnt(0);
#else
      asm volatile("s_wait_tensorcnt 0x0" ::: "memory");
#endif
    }
    __syncthreads();
    const float* fb = &lds_frame[buf][0];   // all further frame reads hit LDS
#else
    const float* fb = fbg;
#endif

    __builtin_prefetch(ff + (size_t)qg * 3, 0, 1);   // global_prefetch_b8

    float best_d[KNN_K]; int best_i[KNN_K];
#pragma unroll
    for (int s = 0; s < KNN_K; ++s) { best_d[s] = 3.0e38f; best_i[s] = 0; }

#ifdef HAVE_WMMA_F32X4
    // --- Build per-step A and C tiles from lane-resident query state -------
    const int s0 = lane & 15;
    const float ax0 = __shfl(qx, s0),      ay0 = __shfl(qy, s0),      az0 = __shfl(qz, s0);
    const float ax1 = __shfl(qx, 16 + s0), ay1 = __shfl(qy, 16 + s0), az1 = __shfl(qz, 16 + s0);
    v2f A0, A1;
    A0.x = hi ? (-2.0f * az0) : (-2.0f * ax0);
    A0.y = hi ? 1.0f          : (-2.0f * ay0);
    A1.x = hi ? (-2.0f * az1) : (-2.0f * ax1);
    A1.y = hi ? 1.0f          : (-2.0f * ay1);
    v8f C0, C1;
#pragma unroll
    for (int r = 0; r < 8; ++r) {
      C0[r] = __shfl(qq, r + (hi ? 8 : 0));
      C1[r] = __shfl(qq, 16 + r + (hi ? 8 : 0));
    }
    float* tb = &lds_tiles[wib][0];
#endif

    // --- Scan all 128 db tiles --------------------------------------------
    for (int tile = 0; tile < N_TILES; ++tile) {
      const int pi = tile * 16 + (lane & 15);
      const float* pp = fb + (size_t)pi * 3;
      const float px = pp[0], py = pp[1], pz = pp[2];

#ifdef HAVE_WMMA_F32X4
      const float pn = px * px + py * py + pz * pz;
      v2f Bm;
      Bm.x = hi ? pz : px;
      Bm.y = hi ? pn : py;
      // D = A x B + C : exact 16x16 squared-distance tiles
      v8f d0 = __builtin_amdgcn_wmma_f32_16x16x4_f32(
          false, A0, false, Bm, (short)0, C0, false, false);
      v8f d1 = __builtin_amdgcn_wmma_f32_16x16x4_f32(
          false, A1, false, Bm, (short)0, C1, false, false);

      // Spill both tiles row-major into this wave's private LDS region.
      const int rb = hi ? 8 : 0;
      const int cb = lane & 15;
#pragma unroll
      for (int r = 0; r < 8; ++r) {
        tb[(rb + r) * 16 + cb]       = d0[r];
        tb[256 + (rb + r) * 16 + cb] = d1[r];
      }
      asm volatile("s_wait_dscnt 0x0" ::: "memory");

      // Lane i consumes the 16 candidates of query i (tile lane>>4, row lane&15),
      // 4 x ds_load_b128 (row base is 64B aligned).
      const float4* rowq =
          (const float4*)(tb + ((lane >> 4) * 256) + ((lane & 15) * 16));
#pragma unroll
      for (int cc = 0; cc < 4; ++cc) {
        const float4 vv = rowq[cc];
        const int dib = tile * 16 + cc * 4;
        KNN_INSERT(vv.x, dib + 0);
        KNN_INSERT(vv.y, dib + 1);
        KNN_INSERT(vv.z, dib + 2);
        KNN_INSERT(vv.w, dib + 3);
      }
#else
      // Fallback: broadcast db points across lanes, per-lane FMA distance.
#pragma unroll
      for (int c = 0; c < 16; ++c) {
        const float bx = __shfl(px, c), by = __shfl(py, c), bz = __shfl(pz, c);
        const float dx = qx - bx, dy = qy - by, dz = qz - bz;
        const float dv = dx * dx + dy * dy + dz * dz;
        KNN_INSERT(dv, tile * 16 + c);
      }
#endif
    }

    // --- Nearest neighbor -> new carry + out_x (reads LDS-staged frame) ---
    const int nn = best_i[0];
    const float nx = fb[(size_t)nn * 3 + 0];
    const float ny = fb[(size_t)nn * 3 + 1];
    const float nz = fb[(size_t)nn * 3 + 2];

    if (dir == 0) {
      const size_t xo = (((size_t)b * T_FRAMES + t_out) * N_PTS + qg) * 3;
      out_outx[xo + 0] = nx; out_outx[xo + 1] = ny; out_outx[xo + 2] = nz;
    }

    // --- Store selected outputs (identity selection of first N/2 queries) -
    if (qg < SEL) {
      const int nout = dir ? (SEL + qg) : qg;
      const size_t obase = (((size_t)b * T_FRAMES + t_out) * N_PTS + nout) * KNN_K;
#pragma unroll 4
      for (int k = 0; k < KNN_K; ++k) {
        const size_t o  = obase + k;
        const int    ii = best_i[k];
        out_dist[o] = best_d[k];
        out_idx0[o] = (float)ii;
        out_idx2[o] = (float)ii;
        const float gx = fb[(size_t)ii * 3 + 0];   // LDS gather
        const float gy = fb[(size_t)ii * 3 + 1];
        const float gz = fb[(size_t)ii * 3 + 2];
        out_pts[o * 3 + 0] = gx; out_pts[o * 3 + 1] = gy; out_pts[o * 3 + 2] = gz;
        const float fx = ff[(size_t)ii * 3 + 0];   // global gather (other frame)
        const float fy = ff[(size_t)ii * 3 + 1];
        const float fz = ff[(size_t)ii * 3 + 2];
        out_feats[o * 6 + 0] = fx; out_feats[o * 6 + 1] = fy; out_feats[o * 6 + 2] = fz;
      }
    }

    qx = nx; qy = ny; qz = nz;
    qq = nx * nx + ny * ny + nz * nz;
  }
}

// ---------------------------------------------------------------------------
// Pass 2: norm = pts - anchor (anchor = pts at segment step 0, k = 0),
// written to both `norm` and feats[..., 3:6]. Pure bandwidth.
// ---------------------------------------------------------------------------
__global__ __launch_bounds__(256)
void norm_fill_kernel(const float* __restrict__ out_pts,
                      float* __restrict__ out_feats,
                      float* __restrict__ out_norm)
{
  const size_t tid   = (size_t)blockIdx.x * blockDim.x + threadIdx.x;
  const size_t total = (size_t)BATCH * T_FRAMES * N_PTS * KNN_K;
  if (tid >= total) return;

  const size_t nk = tid / KNN_K;
  const int n = (int)(nk % N_PTS);
  const size_t tn = nk / N_PTS;
  const int t = (int)(tn % T_FRAMES);
  const int b = (int)(tn / T_FRAMES);
  const int seg = t / STRIDE_T;

  const size_t ao = (((size_t)b * T_FRAMES + seg * STRIDE_T) * N_PTS + n) * KNN_K; // k=0
#pragma unroll
  for (int c = 0; c < 3; ++c) {
    const float a  = out_pts[ao * 3 + c];
    const float p  = out_pts[tid * 3 + c];
    const float nm = p - a;
    out_norm[tid * 3 + c]      = nm;
    out_feats[tid * 6 + 3 + c] = nm;
  }
}

extern "C" void kernel_launch(void* const* d_in, const int* in_sizes, int n_in,
                              void* d_out, int out_size, void* d_ws, size_t ws_size,
                              hipStream_t stream) {
  (void)in_sizes; (void)n_in; (void)out_size; (void)d_ws; (void)ws_size;

  const float* pts_in = (const float*)d_in[0];
  float* out = (float*)d_out;

  const size_t sz = (size_t)BATCH * T_FRAMES * N_PTS * KNN_K;  // 8,388,608
  float* o_idx0 = out;                 // sz
  float* o_dist = out + sz;            // sz
  float* o_idx2 = out + 2 * sz;        // sz
  float* o_pts  = out + 3 * sz;        // 3*sz
  float* o_feat = out + 6 * sz;        // 6*sz
  float* o_norm = out + 12 * sz;       // 3*sz
  float* o_outx = out + 15 * sz;       // B*T*N*3

  // 4096 waves = 512 blocks x 256 threads (8 wave32 per block).
  knn_track_kernel<<<dim3(512), dim3(256), 0, stream>>>(
      pts_in, o_idx0, o_dist, o_idx2, o_pts, o_feat, o_outx);

  const unsigned g2 = (unsigned)((sz + 255) / 256);
  norm_fill_kernel<<<dim3(g2), dim3(256), 0, stream>>>(o_pts, o_feat, o_norm);
}